// SoftMaxFilter2d_24043226923763
// MI455X (gfx1250) — compile-verified
//
#include <hip/hip_runtime.h>
#include <hip/hip_bf16.h>
#include <stdint.h>

// SoftMaxFilter2d: out = boxsum7x7(exp(s*x)*x) / boxsum7x7(exp(s*x)), zero-padded.
// B=8, C=1, H=W=512, kernel 7x7, s=0.1. fp32 in/out.
//
// Memory-bound (~17 MB total traffic -> ~0.72us floor at 23.3 TB/s).
// CDNA5 paths: global_load_async_to_lds_b32 (ASYNCcnt) + s_wait_asynccnt,
// wave32 LDS separable stencil, b128 LDS/global vector accesses,
// odd LDS stride (73) for conflict-free sliding-window reads.
// WMMA deliberately unused: softmax weights are data-dependent (no fixed-
// filter GEMM) and intensity ~4 flop/byte is far below the matrix-core knee.

#define BATCH   8
#define IMG_H   512
#define IMG_W   512
#define RAD     3                    // 7x7 kernel radius
#define KSZ     (2*RAD + 1)          // 7
#define TILE_W  64                   // output tile width
#define TILE_H  32                   // output tile height
#define PAD_W   (TILE_W + 2*RAD)     // 70
#define PAD_H   (TILE_H + 2*RAD)     // 38
#define SV_STR  73                   // odd stride -> adjacent rows on disjoint bank phases
#define TW4     (TILE_W / 4)         // 16 float4 per row
#define NTHREADS 256
#define SHARP   0.1f

__global__ __launch_bounds__(NTHREADS)
void softmax_filter2d_kernel(const float* __restrict__ x, float* __restrict__ out)
{
    // sE first holds the raw staged tile, then is overwritten with e = exp(s*v).
    __shared__ float  sE[PAD_H][SV_STR];
    __shared__ float  sF[PAD_H][SV_STR];
    __shared__ float4 hE[PAD_H][TW4];     // horizontal 7-sums, vectorized
    __shared__ float4 hF[PAD_H][TW4];

    const int tid    = threadIdx.x;
    const int tilesX = IMG_W / TILE_W;          // 8
    const int tilesY = IMG_H / TILE_H;          // 16
    int bid = blockIdx.x;
    const int b  = bid / (tilesX * tilesY);
    int t        = bid % (tilesX * tilesY);
    const int ty = t / tilesX;
    const int tx = t % tilesX;
    const int row0 = ty * TILE_H - RAD;         // padded-tile origin (image coords)
    const int col0 = tx * TILE_W - RAD;

    const float* __restrict__ xb = x + (size_t)b * IMG_H * IMG_W;

    // ---- Stage 1: async-copy padded tile (clamped addresses) into LDS.
    // Tracked by ASYNCcnt; per-lane EXEC masks the partial last iteration.
    for (int i = tid; i < PAD_H * PAD_W; i += NTHREADS) {
        const int pr = i / PAD_W;
        const int pc = i % PAD_W;
        int gr = row0 + pr; gr = gr < 0 ? 0 : (gr >= IMG_H ? IMG_H - 1 : gr);
        int gc = col0 + pc; gc = gc < 0 ? 0 : (gc >= IMG_W ? IMG_W - 1 : gc);
        const unsigned goff    = (unsigned)((gr * IMG_W + gc) * 4);
        const unsigned ldsaddr = (unsigned)(uintptr_t)&sE[pr][pc];
        asm volatile("global_load_async_to_lds_b32 %0, %1, %2"
                     :: "v"(ldsaddr), "v"(goff), "s"(xb)
                     : "memory");
    }
    asm volatile("s_wait_asynccnt 0x0" ::: "memory");  // own wave's async copies done
    __syncthreads();                                   // all waves' copies visible

    // ---- Stage 2: elementwise transform in LDS.
    // Out-of-image elements were clamp-loaded; substitute v=0 (ZeroPad2d):
    // e = exp(0) = 1 joins the denominator, f = 0 joins the numerator.
    // Identical i->thread mapping as stage 1 -> in-place is race-free.
    for (int i = tid; i < PAD_H * PAD_W; i += NTHREADS) {
        const int pr = i / PAD_W;
        const int pc = i % PAD_W;
        const int gr = row0 + pr;
        const int gc = col0 + pc;
        const bool inr = (gr >= 0) && (gr < IMG_H) && (gc >= 0) && (gc < IMG_W);
        const float v = inr ? sE[pr][pc] : 0.0f;
        const float e = __expf(SHARP * v);             // v_exp_f32 path
        sE[pr][pc] = e;
        sF[pr][pc] = e * v;
    }
    __syncthreads();

    // ---- Stage 3: horizontal sliding 7-sums, 4 outputs per thread-step.
    // 38 rows x 16 float4 = 608 vector elements.
    for (int i = tid; i < PAD_H * TW4; i += NTHREADS) {
        const int pr = i >> 4;          // / TW4
        const int c0 = (i & 15) << 2;   // first of 4 output columns
        float we[KSZ + 3], wf[KSZ + 3];
#pragma unroll
        for (int j = 0; j < KSZ + 3; ++j) {            // 10-wide window
            we[j] = sE[pr][c0 + j];
            wf[j] = sF[pr][c0 + j];
        }
        float se = we[0] + we[1] + we[2] + we[3] + we[4] + we[5] + we[6];
        float sf = wf[0] + wf[1] + wf[2] + wf[3] + wf[4] + wf[5] + wf[6];
        float4 ve, vf;
        ve.x = se;                  vf.x = sf;
        se += we[7] - we[0];        sf += wf[7] - wf[0];
        ve.y = se;                  vf.y = sf;
        se += we[8] - we[1];        sf += wf[8] - wf[1];
        ve.z = se;                  vf.z = sf;
        se += we[9] - we[2];        sf += wf[9] - wf[2];
        ve.w = se;                  vf.w = sf;
        hE[pr][i & 15] = ve;        // ds_store_b128, conflict-free
        hF[pr][i & 15] = vf;
    }
    __syncthreads();

    // ---- Stage 4: vertical 7-sums on float4 lanes, divide, b128 store.
    // 32 rows x 16 float4 = 512 vector outputs (2 per thread).
    float* __restrict__ ob = out + (size_t)b * IMG_H * IMG_W;
    for (int i = tid; i < TILE_H * TW4; i += NTHREADS) {
        const int r  = i >> 4;
        const int cq = i & 15;
        float4 ae = make_float4(0.f, 0.f, 0.f, 0.f);
        float4 af = make_float4(0.f, 0.f, 0.f, 0.f);
#pragma unroll
        for (int k = 0; k < KSZ; ++k) {
            const float4 e = hE[r + k][cq];            // ds_load_b128
            const float4 f = hF[r + k][cq];
            ae.x += e.x; ae.y += e.y; ae.z += e.z; ae.w += e.w;
            af.x += f.x; af.y += f.y; af.z += f.z; af.w += f.w;
        }
        float4 o;
        o.x = af.x / ae.x;          // denominator = sum of 49 exps > 0
        o.y = af.y / ae.y;
        o.z = af.z / ae.z;
        o.w = af.w / ae.w;
        const int gr = ty * TILE_H + r;
        const int gc = tx * TILE_W + (cq << 2);
        *(float4*)&ob[gr * IMG_W + gc] = o;            // 16B-aligned global_store_b128
    }
}

extern "C" void kernel_launch(void* const* d_in, const int* in_sizes, int n_in,
                              void* d_out, int out_size, void* d_ws, size_t ws_size,
                              hipStream_t stream)
{
    (void)in_sizes; (void)n_in; (void)d_ws; (void)ws_size; (void)out_size;
    const float* x = (const float*)d_in[0];
    float* out = (float*)d_out;

    const int tiles = (IMG_W / TILE_W) * (IMG_H / TILE_H);   // 128
    dim3 grid(BATCH * tiles);                                 // 1024 blocks
    dim3 block(NTHREADS);
    softmax_filter2d_kernel<<<grid, block, 0, stream>>>(x, out);
}